// GraphTransformer_34110630265293
// MI455X (gfx1250) — compile-verified
//
#include <hip/hip_runtime.h>
#include <math.h>

// ---------------------------------------------------------------------------
// GraphTransformer (3x TransformerConv + MLP) for MI455X / gfx1250 (wave32).
// Dense projections: V_WMMA_F32_16X16X4_F32 (full fp32 matrix pipe), with the
// shared B panel staged into LDS via async global->LDS DMA
// (global_load_async_to_lds_b128 / s_wait_asynccnt). The LDS destination is
// passed to the asm as an addrspace(3) pointer so the array escapes and the
// compiler cannot fold the subsequent ds_loads to undef.
// Edge phase recomputes e = edge_attr @ We on the fly (We is 16KB, L2 resident)
// to avoid materializing a 327MB [E,256] buffer.
// ---------------------------------------------------------------------------

#define N_NODES 20000
#define N_EDGES 320000
#define IN_DIM  128
#define HID     256
#define OUT_DIM 64
#define KSTEP   128          // K-rows staged in LDS per pass (32KB panel)
// HEADS = 4, C = 64 per head, 1/sqrt(C) = 0.125

typedef __attribute__((ext_vector_type(2))) float v2f;
typedef __attribute__((ext_vector_type(8))) float v8f;
typedef __attribute__((address_space(3))) float lds_f32;

__device__ __forceinline__ void atomic_max_float(float* addr, float val) {
    // bit-pattern trick: works for mixed-sign values, init must be -inf
    if (val >= 0.0f) atomicMax((int*)addr, __float_as_int(val));
    else             atomicMin((unsigned int*)addr, __float_as_uint(val));
}

// ---------------------------------------------------------------------------
// Y[M,Nc] = act(X[M,K] @ W[K,Nc] + bias)   (row-major everywhere)
// Block = 256 threads = 8 waves. Each wave owns one 16-row tile; the whole
// block shares one 64-column group of W, staged in LDS in 128-row panels via
// async global->LDS loads. Requires K % 128 == 0, Nc % 64 == 0 (true here).
// All barriers are reached by every wave; compute guards are wave-uniform so
// EXEC is all-ones for every WMMA.
// ---------------------------------------------------------------------------
__global__ void gemm_wmma_f32(const float* __restrict__ X, const float* __restrict__ W,
                              const float* __restrict__ bias, float* __restrict__ Y,
                              int M, int K, int Nc, int do_relu)
{
    __shared__ float Bs[KSTEP * 64];     // 32KB B panel

    const int tid    = threadIdx.x;
    const int lane   = tid & 31;
    const int tile   = (blockIdx.x * blockDim.x + tid) >> 5;   // uniform per wave
    const int mtiles = M >> 4;
    const bool active = (tile < mtiles);

    const int m0   = tile << 4;
    const int n0   = blockIdx.y << 6;
    const int mrow = (active ? m0 : 0) + (lane & 15);
    const int kb   = (lane >> 4) << 1;   // lanes 0-15: K+0/K+1 ; lanes 16-31: K+2/K+3
    const int nlo  = lane & 15;

    v8f acc0 = {}, acc1 = {}, acc2 = {}, acc3 = {};
    const float* xrow = X + (size_t)mrow * K;

    for (int k0 = 0; k0 < K; k0 += KSTEP) {
        // ---- stage W[k0:k0+128, n0:n0+64] -> Bs, async DMA into LDS --------
        // 128 rows x 64 floats = 2048 float4 chunks; 256 threads x 8 iters.
        #pragma unroll
        for (int it = 0; it < (KSTEP * 16) / 256; ++it) {
            const int chunk = it * 256 + tid;        // float4 index
            const int krow  = chunk >> 4;            // 16 chunks per row
            const int joff  = (chunk & 15) << 2;     // float offset in row
            const float* gsrc = W + (size_t)(k0 + krow) * Nc + n0 + joff;
            lds_f32* ldst = (lds_f32*)&Bs[krow * 64 + joff];   // escapes Bs
            asm volatile("global_load_async_to_lds_b128 %0, %1, off"
                         :: "v"(ldst), "v"(gsrc) : "memory");
        }
        asm volatile("s_wait_asynccnt 0x0" ::: "memory");
        __syncthreads();

        if (active) {                                // wave-uniform guard
            // one-step software pipeline on the A fragment
            v2f a;
            a.x = xrow[k0 + kb];
            a.y = xrow[k0 + kb + 1];
            for (int k = 0; k < KSTEP; k += 4) {
                const int kn = (k + 4 < KSTEP) ? (k0 + k + 4) : (k0 + k);
                v2f an;
                an.x = xrow[kn + kb];
                an.y = xrow[kn + kb + 1];
                const float* l0 = &Bs[(k + kb) * 64 + nlo];
                const float* l1 = l0 + 64;
                v2f b0, b1, b2, b3;
                b0.x = l0[0];  b0.y = l1[0];
                b1.x = l0[16]; b1.y = l1[16];
                b2.x = l0[32]; b2.y = l1[32];
                b3.x = l0[48]; b3.y = l1[48];
                acc0 = __builtin_amdgcn_wmma_f32_16x16x4_f32(false, a, false, b0, (short)0, acc0, false, false);
                acc1 = __builtin_amdgcn_wmma_f32_16x16x4_f32(false, a, false, b1, (short)0, acc1, false, false);
                acc2 = __builtin_amdgcn_wmma_f32_16x16x4_f32(false, a, false, b2, (short)0, acc2, false, false);
                acc3 = __builtin_amdgcn_wmma_f32_16x16x4_f32(false, a, false, b3, (short)0, acc3, false, false);
                a = an;
            }
        }
        __syncthreads();                             // panel reuse barrier
    }

    if (!active) return;

    const int rhi = (lane >> 4) << 3;    // C/D: vgpr j -> row j + 8*(lane>=16)
    v8f accs[4] = {acc0, acc1, acc2, acc3};
    #pragma unroll
    for (int t = 0; t < 4; ++t) {
        const int n   = n0 + t * 16 + nlo;
        const float b = bias ? bias[n] : 0.0f;
        #pragma unroll
        for (int j = 0; j < 8; ++j) {
            float v = accs[t][j] + b;
            if (do_relu) v = fmaxf(v, 0.0f);
            Y[(size_t)(m0 + rhi + j) * Nc + n] = v;
        }
    }
}

// ---------------------------------------------------------------------------
// Softmax state init: running max = -inf, denom = 0
// ---------------------------------------------------------------------------
__global__ void init_softmax(float* __restrict__ mmax, float* __restrict__ denom, int n)
{
    int i = blockIdx.x * blockDim.x + threadIdx.x;
    if (i < n) { mmax[i] = __int_as_float(0xFF800000u); denom[i] = 0.0f; }
}

// ---------------------------------------------------------------------------
// Pass 1 per edge (one wave / edge, 8 channels / lane):
//   alpha[e,h] = dot(q[dst], k[src] + edge_attr[e]@We) / 8 ; atomic max per dst
// ---------------------------------------------------------------------------
__global__ void edge_alpha(const float* __restrict__ q, const float* __restrict__ k,
                           const float* __restrict__ ea, const float* __restrict__ We,
                           const int* __restrict__ src, const int* __restrict__ dst,
                           float* __restrict__ alpha, float* __restrict__ mmax)
{
    const int e    = (blockIdx.x * blockDim.x + threadIdx.x) >> 5;
    const int lane = threadIdx.x & 31;
    if (e >= N_EDGES) return;
    const int s = src[e], d = dst[e];

    float ear[16];
    const float* eap = ea + (size_t)e * 16;
    #pragma unroll
    for (int i = 0; i < 16; ++i) ear[i] = eap[i];

    const int c0 = lane * 8;             // lane owns channels [c0, c0+8) -> single head
    const float* qp = q + (size_t)d * HID + c0;
    const float* kp = k + (size_t)s * HID + c0;

    float part = 0.0f;
    #pragma unroll
    for (int c = 0; c < 8; ++c) {
        float ev = 0.0f;
        #pragma unroll
        for (int dd = 0; dd < 16; ++dd) ev = fmaf(ear[dd], We[dd * HID + c0 + c], ev);
        part = fmaf(qp[c], kp[c] + ev, part);
    }
    // reduce the 8 lanes of each head
    part += __shfl_xor(part, 1, 32);
    part += __shfl_xor(part, 2, 32);
    part += __shfl_xor(part, 4, 32);
    if ((lane & 7) == 0) {
        const int h = lane >> 3;
        const float a = part * 0.125f;   // / sqrt(64)
        alpha[(size_t)e * 4 + h] = a;
        atomic_max_float(&mmax[(size_t)d * 4 + h], a);
    }
}

// ---------------------------------------------------------------------------
// Pass 2: ex = exp(alpha - max[dst]); alpha <- ex; denom[dst] += ex
// ---------------------------------------------------------------------------
__global__ void edge_exp(float* __restrict__ alpha, const float* __restrict__ mmax,
                         const int* __restrict__ dst, float* __restrict__ denom)
{
    int i = blockIdx.x * blockDim.x + threadIdx.x;
    if (i >= N_EDGES * 4) return;
    int e = i >> 2, h = i & 3;
    int d = dst[e];
    float ex = expf(alpha[i] - mmax[(size_t)d * 4 + h]);
    alpha[i] = ex;
    atomicAdd(&denom[(size_t)d * 4 + h], ex);
}

// ---------------------------------------------------------------------------
// Pass 3 per edge (one wave / edge): out[dst] += w * (v[src] + e_row)
// out was pre-initialized with the root projection h@Ws+bs.
// ---------------------------------------------------------------------------
__global__ void edge_aggregate(const float* __restrict__ v, const float* __restrict__ ea,
                               const float* __restrict__ We, const int* __restrict__ src,
                               const int* __restrict__ dst, const float* __restrict__ alpha,
                               const float* __restrict__ denom, float* __restrict__ out)
{
    const int e    = (blockIdx.x * blockDim.x + threadIdx.x) >> 5;
    const int lane = threadIdx.x & 31;
    if (e >= N_EDGES) return;
    const int s = src[e], d = dst[e];
    const int h = lane >> 3;
    const float w = alpha[(size_t)e * 4 + h] / (denom[(size_t)d * 4 + h] + 1e-16f);

    float ear[16];
    const float* eap = ea + (size_t)e * 16;
    #pragma unroll
    for (int i = 0; i < 16; ++i) ear[i] = eap[i];

    const int c0 = lane * 8;
    const float* vp = v   + (size_t)s * HID + c0;
    float*       op = out + (size_t)d * HID + c0;
    #pragma unroll
    for (int c = 0; c < 8; ++c) {
        float ev = 0.0f;
        #pragma unroll
        for (int dd = 0; dd < 16; ++dd) ev = fmaf(ear[dd], We[dd * HID + c0 + c], ev);
        atomicAdd(&op[c], w * (vp[c] + ev));
    }
}

// ---------------------------------------------------------------------------
__global__ void relu_copy(const float* __restrict__ in, float* __restrict__ out, int n)
{
    int i = blockIdx.x * blockDim.x + threadIdx.x;
    if (i < n) out[i] = fmaxf(in[i], 0.0f);
}

// ---------------------------------------------------------------------------
extern "C" void kernel_launch(void* const* d_in, const int* in_sizes, int n_in,
                              void* d_out, int out_size, void* d_ws, size_t ws_size,
                              hipStream_t stream)
{
    // pytree insertion order:
    // 0:x 1:edge_index 2:batch 3:edge_attr
    // layer l at 4+9l: Wq,bq,Wk,bk,Wv,bv,We,Ws,bs    (l = 0,1,2)
    // 31:W1 32:b1 33:W2 34:b2
    const float* x  = (const float*)d_in[0];
    const int*   ei = (const int*)d_in[1];
    const int*   src = ei;
    const int*   dst = ei + N_EDGES;
    const float* ea = (const float*)d_in[3];
    auto P = [&](int i) { return (const float*)d_in[i]; };

    // workspace carve-up (floats): ~129 MB total
    float* ws = (float*)d_ws;
    const size_t NH = (size_t)N_NODES * HID;          // 5.12M floats
    float* hA    = ws;
    float* hB    = ws + NH;
    float* qb    = ws + 2 * NH;
    float* kbuf  = ws + 3 * NH;
    float* vbuf  = ws + 4 * NH;
    float* accum = ws + 5 * NH;
    float* alpha = ws + 6 * NH;                       // E*4 floats
    float* mmax  = alpha + (size_t)N_EDGES * 4;       // N*4
    float* denom = mmax + (size_t)N_NODES * 4;        // N*4

    auto gemm = [&](const float* X, const float* W, const float* bias, float* Y,
                    int M, int K, int Nc, int relu) {
        int mtiles = M / 16;
        dim3 grid((mtiles + 7) / 8, Nc / 64);         // 8 waves per 256-thread block
        gemm_wmma_f32<<<grid, 256, 0, stream>>>(X, W, bias, Y, M, K, Nc, relu);
    };

    const int edge_wave_blocks = (N_EDGES * 32) / 256;        // 40000
    const int eh_blocks        = (N_EDGES * 4 + 255) / 256;   // 5000
    const int nh_blocks        = (int)((NH + 255) / 256);     // 20000
    const int nm_blocks        = (N_NODES * 4 + 255) / 256;

    const float* hin = x;
    int K = IN_DIM;
    float* houts[3] = {hA, hB, hA};

    for (int l = 0; l < 3; ++l) {
        const int base = 4 + 9 * l;
        const float *Wq = P(base + 0), *bq = P(base + 1);
        const float *Wk = P(base + 2), *bk = P(base + 3);
        const float *Wv = P(base + 4), *bv = P(base + 5);
        const float *We = P(base + 6);
        const float *Ws = P(base + 7), *bs = P(base + 8);

        gemm(hin, Wq, bq, qb,    N_NODES, K, HID, 0);
        gemm(hin, Wk, bk, kbuf,  N_NODES, K, HID, 0);
        gemm(hin, Wv, bv, vbuf,  N_NODES, K, HID, 0);
        gemm(hin, Ws, bs, accum, N_NODES, K, HID, 0);   // root/skip term -> accumulator

        init_softmax<<<nm_blocks, 256, 0, stream>>>(mmax, denom, N_NODES * 4);
        edge_alpha<<<edge_wave_blocks, 256, 0, stream>>>(qb, kbuf, ea, We, src, dst, alpha, mmax);
        edge_exp<<<eh_blocks, 256, 0, stream>>>(alpha, mmax, dst, denom);
        edge_aggregate<<<edge_wave_blocks, 256, 0, stream>>>(vbuf, ea, We, src, dst, alpha, denom, accum);

        float* hout = houts[l];
        relu_copy<<<nh_blocks, 256, 0, stream>>>(accum, hout, (int)NH);
        hin = hout;
        K = HID;
    }

    // MLP head: relu(h @ W1 + b1) @ W2 + b2
    gemm(houts[2], P(31), P(32), qb, N_NODES, HID, HID, 1);
    gemm(qb, P(33), P(34), (float*)d_out, N_NODES, HID, OUT_DIM, 0);
}